// VoxelGrid_62354335203431
// MI455X (gfx1250) — compile-verified
//
#include <hip/hip_runtime.h>

// ---- problem constants (must match reference) ----
#define XMINF (-51.2f)
#define YMINF (-51.2f)
#define ZMINF (-3.0f)
#define NXc 205          // int((51.2 - -51.2)/0.5) + 1
#define NYc 205
#define NZc 13           // int((3.0 - -3.0)/0.5) + 1
#define INV_VOXEL 2.0f   // 1/0.5 exactly

typedef __attribute__((ext_vector_type(4))) float v4f;

__device__ __forceinline__ void tri1(float px, float py, float pz,
                                     const float* __restrict__ g,
                                     float& o0, float& o1, float& o2) {
  // grid coords (exact: /0.5 == *2)
  float gx = (px - XMINF) * INV_VOXEL;
  float gy = (py - YMINF) * INV_VOXEL;
  float gz = (pz - ZMINF) * INV_VOXEL;
  int fx = (int)floorf(gx);
  int fy = (int)floorf(gy);
  int fz = (int)floorf(gz);
  int x0 = min(max(fx, 0), NXc - 1);
  int y0 = min(max(fy, 0), NYc - 1);
  int z0 = min(max(fz, 0), NZc - 1);
  int x1 = min(max(fx + 1, 0), NXc - 1);
  int y1 = min(max(fy + 1, 0), NYc - 1);
  int z1 = min(max(fz + 1, 0), NZc - 1);
  float xd = gx - (float)x0;
  float yd = gy - (float)y0;
  float zd = gz - (float)z0;

  const int SX = NYc * NZc * 3;  // x stride in floats
  const int SY = NZc * 3;        // y stride in floats
  int bx0 = x0 * SX, bx1 = x1 * SX;
  int by0 = y0 * SY, by1 = y1 * SY;
  int bz0 = z0 * 3,  bz1 = z1 * 3;

  int b000 = bx0 + by0 + bz0;
  int b001 = bx0 + by0 + bz1;
  int b010 = bx0 + by1 + bz0;
  int b011 = bx0 + by1 + bz1;
  int b100 = bx1 + by0 + bz0;
  int b101 = bx1 + by0 + bz1;
  int b110 = bx1 + by1 + bz0;
  int b111 = bx1 + by1 + bz1;

  float omx = 1.0f - xd, omy = 1.0f - yd, omz = 1.0f - zd;

#pragma unroll
  for (int ch = 0; ch < 3; ++ch) {
    // grid gathers: default (RT) temporal policy -> stays hot in WGP$/L2
    float c000 = g[b000 + ch];
    float c001 = g[b001 + ch];
    float c010 = g[b010 + ch];
    float c011 = g[b011 + ch];
    float c100 = g[b100 + ch];
    float c101 = g[b101 + ch];
    float c110 = g[b110 + ch];
    float c111 = g[b111 + ch];
    float c00 = c000 * omx + c100 * xd;
    float c01 = c001 * omx + c101 * xd;
    float c10 = c010 * omx + c110 * xd;
    float c11 = c011 * omx + c111 * xd;
    float c0 = c00 * omy + c10 * yd;
    float c1 = c01 * omy + c11 * yd;
    float r = c0 * omz + c1 * zd;
    if (ch == 0) o0 = r;
    else if (ch == 1) o1 = r;
    else o2 = r;
  }
}

// One thread handles 4 points: 3x b128 NT loads in, 3x b128 NT stores out.
__global__ __launch_bounds__(256)
void voxel_trilerp_kernel(const float* __restrict__ points,
                          const float* __restrict__ grid,
                          float* __restrict__ out, int npts) {
  int t = blockIdx.x * blockDim.x + threadIdx.x;
  int p0 = t * 4;
  if (p0 + 3 < npts || (p0 + 4 == npts + (npts & 3)) ) {} // (no-op; clarity below)

  if (p0 + 3 < npts) {
    const v4f* pin = (const v4f*)points;
    // Prefetch the point stream ~768 KB ahead into L2 (speculative; OOB dropped).
    __builtin_prefetch(points + (size_t)12 * t + 196608, 0, 1);

    v4f a = __builtin_nontemporal_load(pin + 3 * t);
    v4f b = __builtin_nontemporal_load(pin + 3 * t + 1);
    v4f c = __builtin_nontemporal_load(pin + 3 * t + 2);

    float r0x, r0y, r0z, r1x, r1y, r1z, r2x, r2y, r2z, r3x, r3y, r3z;
    tri1(a.x, a.y, a.z, grid, r0x, r0y, r0z);
    tri1(a.w, b.x, b.y, grid, r1x, r1y, r1z);
    tri1(b.z, b.w, c.x, grid, r2x, r2y, r2z);
    tri1(c.y, c.z, c.w, grid, r3x, r3y, r3z);

    v4f o0 = {r0x, r0y, r0z, r1x};
    v4f o1 = {r1y, r1z, r2x, r2y};
    v4f o2 = {r2z, r3x, r3y, r3z};
    v4f* pout = (v4f*)out;
    __builtin_nontemporal_store(o0, pout + 3 * t);
    __builtin_nontemporal_store(o1, pout + 3 * t + 1);
    __builtin_nontemporal_store(o2, pout + 3 * t + 2);
  } else if (p0 < npts) {
    // ragged tail (not hit for 8M points, but keeps the kernel general)
    for (int p = p0; p < npts; ++p) {
      float px = points[3 * p + 0];
      float py = points[3 * p + 1];
      float pz = points[3 * p + 2];
      float ox, oy, oz;
      tri1(px, py, pz, grid, ox, oy, oz);
      out[3 * p + 0] = ox;
      out[3 * p + 1] = oy;
      out[3 * p + 2] = oz;
    }
  }
}

extern "C" void kernel_launch(void* const* d_in, const int* in_sizes, int n_in,
                              void* d_out, int out_size, void* d_ws, size_t ws_size,
                              hipStream_t stream) {
  const float* points = (const float*)d_in[0];  // (N_POINTS, 3) f32
  const float* grid   = (const float*)d_in[1];  // (205, 205, 13, 3) f32
  float* out = (float*)d_out;                   // (N_POINTS, 3) f32

  int npts = in_sizes[0] / 3;
  int nthreads = (npts + 3) / 4;
  int blocks = (nthreads + 255) / 256;
  hipLaunchKernelGGL(voxel_trilerp_kernel, dim3(blocks), dim3(256), 0, stream,
                     points, grid, out, npts);
}